// photoreceptor_RODS_REIKE_14929306321260
// MI455X (gfx1250) — compile-verified
//
#include <hip/hip_runtime.h>
#include <stdint.h>

// Problem shape fixed by the reference: X is [B, T, P] float32.
#define B_DIM 16
#define T_DIM 1024
#define P_DIM 2048

#define BP 128                         // p-columns per block (one block per (b, p-tile))
#define CT 32                          // timesteps staged per chunk
#define NWAVES (BP / 32)               // 4 wave32 per block
#define ROWS_PW (CT / NWAVES)          // 8 rows DMA'd per wave per chunk
#define NCHUNK ((T_DIM - 1 + CT - 1) / CT)      // 32 chunks cover X rows 0..1022 (+1 harmless row)
#define V4_PER_THREAD ((CT * BP / 4) / BP)      // 8 float4 async loads per thread (fallback path)

// ---- staging path selection: TDM > async global->LDS > synchronous ----
#if defined(__gfx1250__) && __has_builtin(__builtin_amdgcn_tensor_load_to_lds)
#define HAVE_TDM 1
#else
#define HAVE_TDM 0
#endif
#if defined(__gfx1250__) && __has_builtin(__builtin_amdgcn_global_load_async_to_lds_b128)
#define HAVE_ASYNC 1
#else
#define HAVE_ASYNC 0
#endif

#if HAVE_TDM
#define WAIT_STAGE(n) __builtin_amdgcn_s_wait_tensorcnt(n)
#define STAGE_INFLIGHT 1
#elif HAVE_ASYNC
#if __has_builtin(__builtin_amdgcn_s_wait_asynccnt)
#define WAIT_STAGE(n) __builtin_amdgcn_s_wait_asynccnt(n)
#else
#define WAIT_STAGE(n) asm volatile("s_wait_asynccnt %0" ::"i"(n) : "memory")
#endif
#define STAGE_INFLIGHT V4_PER_THREAD
#else
#define WAIT_STAGE(n)
#define STAGE_INFLIGHT 0
#endif

typedef unsigned int u32x4 __attribute__((ext_vector_type(4)));
typedef int i32x8 __attribute__((ext_vector_type(8)));
typedef int i32x4 __attribute__((ext_vector_type(4)));
typedef int v4i __attribute__((ext_vector_type(4)));
typedef __attribute__((address_space(1))) v4i* gptr_b128;
typedef __attribute__((address_space(3))) v4i* lptr_b128;
#define AS1_B128(p) ((gptr_b128)(uint64_t)(uintptr_t)(p))
#define AS3_B128(p) ((lptr_b128)(uint32_t)(uintptr_t)(p))

__global__ __launch_bounds__(BP) void rieke_scan_kernel(
    const float* __restrict__ X,
    const float* __restrict__ sig_raw,
    const float* __restrict__ phi_raw,
    const float* __restrict__ eta_raw,
    const float* __restrict__ beta_raw,
    float* __restrict__ out)
{
    __shared__ float buf[2][CT][BP];

    const int tid = threadIdx.x;
    const int pblocks = P_DIM / BP;                 // 16
    const int b  = blockIdx.x / pblocks;
    const int p0 = (blockIdx.x % pblocks) * BP;

    // --- trainable params with scale factors ---
    const float sigma = sig_raw[0] * 10.0f;
    const float phi   = phi_raw[0] * 10.0f;
    const float eta   = eta_raw[0] * 100.0f;
    const float beta  = beta_raw[0] * 100.0f;

    // --- fixed biophysics; exponents are exact small integers ---
    const float dt     = 0.008f;
    const float gamma_ = 12.5f;
    const float gdark  = 20.0f;
    const float darkI  = 40.0f;                     // 20^3 * 0.01 / 2
    const float cur2ca = beta / darkI;              // beta * CDARK / darkI
    const float smax   = (eta / phi) * gdark * 17.0f;   // 1 + (1/0.5)^4 = 17
    const float kr = 1.0f - dt * sigma;             // r decay per step
    const float k1 = dt * cur2ca * 0.005f;          // dt * cur2ca * CGMP2CUR/2
    const float k2 = dt * beta;

    const size_t base = ((size_t)b * T_DIM) * P_DIM + p0;
    const float* Xb = X + base;
    float* Ob = out + base;

    // --- initial state (uses X[b,0,p]) ---
    const float x0 = Xb[tid];
    float rv  = x0 * gamma_ / sigma;
    float pv  = (eta + rv) / phi;
    float cst = 1.0f;                               // CDARK
    float sv  = gdark * eta / phi;
    float gv  = gdark;

    // out[b,0,p] = -(0.01 * 0^3)/2 = -0.0
    __builtin_nontemporal_store(-0.0f, Ob + tid);

#if HAVE_TDM
    // Each wave DMAs its own ROWS_PW x BP sub-tile of the chunk via the
    // Tensor Data Mover. Descriptors are wave-uniform; EXEC is ignored by
    // TDM ops, which is safe here because execution is convergent.
    const unsigned wid = (unsigned)__builtin_amdgcn_readfirstlane(tid >> 5);
    const uint32_t lds0 = (uint32_t)(uintptr_t)&buf[0][0][0];
    auto issue = [&](int k, int bs) {
        const uint64_t ga = (uint64_t)(uintptr_t)(Xb + (size_t)(k * CT + wid * ROWS_PW) * P_DIM);
        const uint32_t la = lds0 + (uint32_t)bs * (CT * BP * 4u) + wid * (ROWS_PW * BP * 4u);
        // D# group 0: count=1 | lds_addr | global_addr[56:0] | type=2
        u32x4 g0 = { 1u, la, (uint32_t)ga,
                     (((uint32_t)(ga >> 32)) & 0x01FFFFFFu) | 0x80000000u };
        // D# group 1: data_size=2 (4B); tensor_dim0=BP; tensor_dim1=ROWS_PW;
        // tile_dim0=BP; tile_dim1=ROWS_PW; tensor_dim0_stride=P_DIM.
        i32x8 g1 = { (int)0x00020000u,
                     (int)((unsigned)BP << 16),        // tensor_dim0[15:0] @ bits 63:48
                     (int)((unsigned)ROWS_PW << 16),   // tensor_dim1[15:0] @ bits 95:80
                     (int)((unsigned)BP << 16),        // tile_dim0 @ bits 127:112
                     (int)ROWS_PW,                     // tile_dim1 @ bits 143:128
                     (int)P_DIM,                       // tensor_dim0_stride[31:0]
                     0, 0 };
        i32x4 z4 = { 0, 0, 0, 0 };                     // 2D tile: groups 2/3 unused
        i32x8 z8 = { 0, 0, 0, 0, 0, 0, 0, 0 };
        __builtin_amdgcn_tensor_load_to_lds(g0, g1, z4, z4, z8, 0);
    };
#else
    // Fallback: per-thread staging of chunk k into buf[bs].
    auto issue = [&](int k, int bs) {
#pragma unroll
        for (int i = 0; i < V4_PER_THREAD; ++i) {
            const int f   = tid + i * BP;           // float4 slot within chunk
            const int row = f >> 5;                 // BP/4 = 32 vec4 per row
            const int cvc = f & 31;
            const float* gsrc = Xb + (size_t)(k * CT + row) * P_DIM + cvc * 4;
            float* ldst = &buf[bs][row][cvc * 4];
#if HAVE_ASYNC
            __builtin_amdgcn_global_load_async_to_lds_b128(AS1_B128(gsrc), AS3_B128(ldst), 0, 0);
#else
            const float4 v = *(const float4*)gsrc;
            *(float4*)ldst = v;
#endif
        }
    };
#endif

    int sel = 0;
    issue(0, 0);
    for (int k = 0; k < NCHUNK; ++k) {
        if (k + 1 < NCHUNK) {
            issue(k + 1, sel ^ 1);                  // prefetch next chunk
            WAIT_STAGE(STAGE_INFLIGHT);             // in-order retire: chunk k landed
        } else {
            WAIT_STAGE(0);
        }
        __syncthreads();                            // all waves' sub-tiles of chunk k visible

        const int rows = ((T_DIM - 1) - k * CT) < CT ? ((T_DIM - 1) - k * CT) : CT;
        const float* __restrict__ lbuf = &buf[sel][0][0];
        float* __restrict__ obase = Ob + (size_t)(k * CT + 1) * P_DIM + tid;

#pragma unroll 4
        for (int i = 0; i < rows; ++i) {
            const float x  = lbuf[i * BP + tid];
            // carry update (all RHS use old state except s_n which uses c_n)
            const float rn = fmaf(rv, kr, gamma_ * x);
            const float pn = fmaf(dt, (rv + eta) - phi * pv, pv);
            const float g3 = gv * gv * gv;
            const float cn = fmaf(k1, g3, fmaf(-k2, cst, cst));
            const float u  = 2.0f * cn;             // c_n / 0.5
            const float u2 = u * u;
            const float sn = smax * __builtin_amdgcn_rcpf(fmaf(u2, u2, 1.0f));
            const float gn = fmaf(dt, fmaf(-pv, gv, sv), gv);
            rv = rn; pv = pn; cst = cn; sv = sn; gv = gn;

            const float o = -0.005f * (gn * gn * gn);   // -(CGMP2CUR * g^3)/2
            __builtin_nontemporal_store(o, obase + (size_t)i * P_DIM);
        }
        __syncthreads();                            // buf[sel] free for reuse
        sel ^= 1;
    }
}

extern "C" void kernel_launch(void* const* d_in, const int* in_sizes, int n_in,
                              void* d_out, int out_size, void* d_ws, size_t ws_size,
                              hipStream_t stream) {
    const float* X    = (const float*)d_in[0];
    const float* sig  = (const float*)d_in[1];
    const float* phi  = (const float*)d_in[2];
    const float* eta  = (const float*)d_in[3];
    const float* beta = (const float*)d_in[4];
    float* out = (float*)d_out;

    dim3 grid(B_DIM * (P_DIM / BP));   // 256 blocks
    dim3 block(BP);                    // 128 threads = 4 wave32
    hipLaunchKernelGGL(rieke_scan_kernel, grid, block, 0, stream,
                       X, sig, phi, eta, beta, out);
}